// DRCLayer_37495064494841
// MI455X (gfx1250) — compile-verified
//
#include <hip/hip_runtime.h>
#include <math.h>

#define CHUNK 4096
#define NT 256

__device__ __forceinline__ float readlane_f(float v, int k) {
    return __int_as_float(__builtin_amdgcn_readlane(__float_as_int(v), k));
}

__device__ __forceinline__ float sanitize_p(float p) {
    p = (p != p) ? 0.0f : p;          // NaN -> 0
    p = (p == 0.0f) ? 1e-10f : p;     // 0 -> 1e-10
    return p;
}

// x_L = x_G - y_G  (level detector + soft-knee static curve)
__device__ __forceinline__ float comp_xl(float xv, float T, float invR,
                                         float knee, float inv2knee) {
    // 20*log10(v) = 6.020599913 * log2(v)
    float xG = fmaxf(6.0205999132796239f * __log2f(fabsf(xv) + 1e-8f), -96.0f);
    float t  = xG - T;
    float t2 = 2.0f * t;
    float yG;
    if (t2 < -knee) {
        yG = xG;
    } else if (t2 > knee) {
        yG = T + t * invR;
    } else {
        float h = (t + knee) * 0.5f;
        yG = xG + invR * h * h * inv2knee;
    }
    return xG - yG;
}

__device__ __forceinline__ float finish_sample(float xv, float z, float sgn, float mk) {
    float yL = fmaxf(sgn * z, -96.0f);
    // 10^(d/20) = 2^(d * log2(10)/20)
    float g  = exp2f((mk - yL) * 0.16609640474436813f);
    float o  = xv * g;
    // scrub non-finite (NaN/Inf compare false)
    o = (fabsf(o) <= 3.402823466e+38f) ? o : 0.0f;
    return o;
}

extern "C" __global__ __launch_bounds__(NT, 1)
void drc_scan_kernel(const float* __restrict__ x,
                     const float* __restrict__ params,
                     float* __restrict__ out, int N) {
    __shared__ __align__(16) float xbuf[2][CHUNK];   // raw input staged via async engine
    __shared__ __align__(16) float xlbuf[2][CHUNK];  // x_L, overwritten in place with scan state z

    const int row  = blockIdx.x;
    const int tid  = threadIdx.x;
    const int lane = tid & 31;
    const int wid  = tid >> 5;

    // ---- per-row uniforms (scalar path; redundant per-thread is fine) ----
    const float* pr = params + row * 6;
    const float p0 = sanitize_p(pr[0]);
    const float p1 = sanitize_p(pr[1]);
    const float p2 = sanitize_p(pr[2]);
    const float p3 = sanitize_p(pr[3]);
    const float p4 = sanitize_p(pr[4]);
    const float p5 = sanitize_p(pr[5]);

    const float T     = -p0 * 60.0f;
    const float ratio =  p1 * 10.0f;
    const float atk   = fmaxf(p2 * 0.1f, 1e-4f);
    const float rel   = fmaxf(p3 * 3.0f, 0.005f);
    const float knee  = p4 * 24.0f;
    const float mk    = p5 * 20.0f;

    const float LOG9 = 2.1972245773362196f;
    const float aA = expf(-LOG9 / (44100.0f * atk));
    const float aR = expf(-LOG9 / (44100.0f * rel));
    const float invR     = 1.0f / ratio;
    const float inv2knee = 1.0f / (2.0f * knee);

    // Branching smoother as max-of-affine:
    //   y' = max(aA*y + (1-aA)*xl, aR*y + (1-aR)*xl)  iff aA <= aR, else min.
    // Fold min-case into max via z = sgn*y.
    const float sgn = (aA <= aR) ? 1.0f : -1.0f;
    const float bAs = sgn * (1.0f - aA);
    const float bRs = sgn * (1.0f - aR);

    const float* xr  = x   + (size_t)row * (size_t)N;
    float*       our = out + (size_t)row * (size_t)N;
    const unsigned long long xbase = (unsigned long long)(uintptr_t)xr;

    const int nc = (N + CHUNK - 1) / CHUNK;

    // ---- producer: async global->LDS copy of x, then compute x_L into LDS ----
    auto produce = [&](int c, int tfirst, int nthr) {
        const int cbase = c * CHUNK;
        const int cnt   = min(CHUNK, N - cbase);
        const int q     = cnt >> 2;   // float4 groups
        const int b     = c & 1;
        const int t     = tid - tfirst;
        // CDNA5 async-tensor path: each lane moves 16B global->LDS (ASYNCcnt tracked)
        for (int u = t; u < q; u += nthr) {
            unsigned int ldsOff = (unsigned int)(uintptr_t)(&xbuf[b][u << 2]);
            unsigned int gOff   = (unsigned int)((cbase + (u << 2)) * 4);
            asm volatile("global_load_async_to_lds_b128 %0, %1, %2"
                         :: "v"(ldsOff), "v"(gOff), "s"(xbase) : "memory");
        }
        asm volatile("s_wait_asynccnt 0" ::: "memory");
        for (int u = t; u < q; u += nthr) {
            const float4 xv = *(const float4*)(&xbuf[b][u << 2]);
            float4 r;
            r.x = comp_xl(xv.x, T, invR, knee, inv2knee);
            r.y = comp_xl(xv.y, T, invR, knee, inv2knee);
            r.z = comp_xl(xv.z, T, invR, knee, inv2knee);
            r.w = comp_xl(xv.w, T, invR, knee, inv2knee);
            *(float4*)(&xlbuf[b][u << 2]) = r;
        }
        for (int u = (q << 2) + t; u < cnt; u += nthr) {  // scalar tail (N % 4)
            float xv = xr[cbase + u];
            xbuf[b][u]  = xv;
            xlbuf[b][u] = comp_xl(xv, T, invR, knee, inv2knee);
        }
    };

    // ---- consumer scan state: z = sgn * y, persists across chunks in wave 0 ----
    float z = 0.0f;

    auto scan = [&](int c) {
        const int cnt = min(CHUNK, N - c * CHUNK);
        const int b   = c & 1;
        int g = 0;
        for (; g + 32 <= cnt; g += 32) {
            const float xl = xlbuf[b][g + lane];
            const float vA = bAs * xl;   // off critical path
            const float vR = bRs * xl;
            float zrec = 0.0f;
            #pragma unroll
            for (int k = 0; k < 32; ++k) {
                const float sA = readlane_f(vA, k);
                const float sR = readlane_f(vR, k);
                z = fmaxf(fmaf(aA, z, sA), fmaf(aR, z, sR));  // chain: fma -> max
                zrec = (lane == k) ? z : zrec;
            }
            xlbuf[b][g + lane] = zrec;  // overwrite x_L with scan state
        }
        if (g < cnt) {  // guarded tail group (cnt % 32)
            const int m = cnt - g;
            const float xl = (lane < m) ? xlbuf[b][g + lane] : 0.0f;
            const float vA = bAs * xl;
            const float vR = bRs * xl;
            float zrec = 0.0f;
            #pragma unroll
            for (int k = 0; k < 32; ++k) {
                const float sA = readlane_f(vA, k);
                const float sR = readlane_f(vR, k);
                const float zn = fmaxf(fmaf(aA, z, sA), fmaf(aR, z, sR));
                z = (k < m) ? zn : z;
                zrec = (lane == k) ? z : zrec;
            }
            if (lane < m) xlbuf[b][g + lane] = zrec;
        }
    };

    // ---- emit: gain + scrub + vectorized store ----
    auto emit = [&](int c) {
        const int cbase = c * CHUNK;
        const int cnt   = min(CHUNK, N - cbase);
        const int q     = cnt >> 2;
        const int b     = c & 1;
        for (int u = tid; u < q; u += NT) {
            const float4 xv = *(const float4*)(&xbuf[b][u << 2]);
            const float4 zv = *(const float4*)(&xlbuf[b][u << 2]);
            float4 o;
            o.x = finish_sample(xv.x, zv.x, sgn, mk);
            o.y = finish_sample(xv.y, zv.y, sgn, mk);
            o.z = finish_sample(xv.z, zv.z, sgn, mk);
            o.w = finish_sample(xv.w, zv.w, sgn, mk);
            *(float4*)(our + cbase + (u << 2)) = o;
        }
        for (int u = (q << 2) + tid; u < cnt; u += NT) {
            our[cbase + u] = finish_sample(xbuf[b][u], xlbuf[b][u], sgn, mk);
        }
    };

    // ---- pipeline: prologue fills chunk 0, then scan(c) overlaps produce(c+1) ----
    produce(0, 0, NT);
    if (tid == 0) xlbuf[0][0] = 0.0f;  // y_L[0] = 0 and state stays 0 (exact)
    __syncthreads();

    for (int c = 0; c < nc; ++c) {
        if (wid == 0) {
            scan(c);                       // serial recurrence (critical path)
        } else if (c + 1 < nc) {
            produce(c + 1, 32, NT - 32);   // hidden behind the scan
        }
        __syncthreads();
        emit(c);
        __syncthreads();
    }
}

extern "C" void kernel_launch(void* const* d_in, const int* in_sizes, int n_in,
                              void* d_out, int out_size, void* d_ws, size_t ws_size,
                              hipStream_t stream) {
    (void)n_in; (void)d_ws; (void)ws_size; (void)out_size;
    const float* x = (const float*)d_in[0];
    const float* p = (const float*)d_in[1];
    float* out = (float*)d_out;
    const int B = in_sizes[1] / 6;          // params is (B, 6)
    const int N = (B > 0) ? (in_sizes[0] / B) : 0;
    if (B <= 0 || N <= 0) return;
    drc_scan_kernel<<<B, NT, 0, stream>>>(x, p, out, N);
}